// PatchTransformerKeypoints_72301479461243
// MI455X (gfx1250) — compile-verified
//
#include <hip/hip_runtime.h>
#include <stdint.h>

#define PDIM 300
#define KPAD 3
#define CROP_X0 50
#define CROP_H 200
#define CROP_W 300
#define SIMG 416

// ---------------------------------------------------------------------------
// Kernel 1: 7x7 median pool with reflect padding.
// One thread per output pixel; 49 values held in VGPRs; partial selection
// network (25 fully-unrolled min/max sweeps) -> element of rank 24.
// ---------------------------------------------------------------------------
__global__ __launch_bounds__(256)
void median_pool_kernel(const float* __restrict__ in, float* __restrict__ out) {
    int idx = blockIdx.x * blockDim.x + threadIdx.x;
    if (idx >= 3 * PDIM * PDIM) return;
    int c   = idx / (PDIM * PDIM);
    int rem = idx - c * (PDIM * PDIM);
    int y   = rem / PDIM;
    int x   = rem - y * PDIM;
    const float* plane = in + c * PDIM * PDIM;

    float w[49];
    #pragma unroll
    for (int dy = 0; dy < 7; ++dy) {
        int ty = y + dy - KPAD;
        ty = (ty < 0) ? -ty : ty;                      // reflect (no edge repeat)
        ty = (ty >= PDIM) ? (2 * PDIM - 2 - ty) : ty;
        #pragma unroll
        for (int dx = 0; dx < 7; ++dx) {
            int tx = x + dx - KPAD;
            tx = (tx < 0) ? -tx : tx;
            tx = (tx >= PDIM) ? (2 * PDIM - 2 - tx) : tx;
            w[dy * 7 + dx] = plane[ty * PDIM + tx];
        }
    }
    // Selection: after pass i, w[i] holds the i-th smallest. Need rank 24.
    #pragma unroll
    for (int i = 0; i <= 24; ++i) {
        #pragma unroll
        for (int j = i + 1; j < 49; ++j) {
            float lo = fminf(w[i], w[j]);
            float hi = fmaxf(w[i], w[j]);
            w[i] = lo;
            w[j] = hi;
        }
    }
    out[idx] = w[24];
}

// ---------------------------------------------------------------------------
// Kernel 2: per-box bilinear resample + in-box masking.
// grid = (64 boxes, 416 rows); block = 416 threads (one per x pixel).
// Active rows stage the two needed pooled rows (cols 50..299, 3 channels)
// into LDS via CDNA5 async global->LDS loads (ASYNCcnt / s_wait_asynccnt),
// then all bilinear taps are served from LDS.
// ---------------------------------------------------------------------------
__global__ __launch_bounds__(416)
void patch_apply_kernel(const float* __restrict__ pooled,
                        const float* __restrict__ lab,
                        const int*   __restrict__ img_size_p,
                        float* __restrict__ out) {
    const int box = blockIdx.x;   // b*8 + n, 0..63
    const int y   = blockIdx.y;   // 0..415
    const int tid = threadIdx.x;  // 0..415

    // --- block-uniform box parameters (scalar loads) ---
    const float* kp = lab + box * 39;
    float k15 = kp[15], k16 = kp[16], k17 = kp[17], k18 = kp[18];
    float k20 = kp[20], k37i = kp[37], k38 = kp[38];
    float fimg = (float)(*img_size_p);
    float k37 = (k38 == 0.0f) ? fminf(fabsf(k18 - k15) * 1.7f + 1.0f, fimg) : k37i;
    float Ht = floorf(fabsf(k37 - k16) * 1.2f) + 1.0f;
    float Wt = floorf(fabsf(k18 - k15) * 1.3f) + 1.0f;
    float L  = floorf(fminf(k15, k18));
    float T  = floorf(fminf(k16, k37));
    bool cond = (k17 > 0.0f) && (k20 > 0.0f);

    float yf = (float)y;
    bool row_active = cond && (yf >= T) && (yf < T + Ht);

    float* orow = out + (size_t)box * 3u * (SIMG * SIMG) + (size_t)y * SIMG;

    if (!row_active) {   // uniform per block: whole row is masked to zero
        orow[tid]                         = 0.0f;
        orow[1u * SIMG * SIMG + tid]      = 0.0f;
        orow[2u * SIMG * SIMG + tid]      = 0.0f;
        return;
    }

    // --- vertical sample coords (uniform across block) ---
    float v   = (yf - T + 0.5f) * ((float)CROP_H / Ht) - 0.5f;
    float y0f = floorf(v);
    float wy  = v - y0f;
    int y0 = (int)fminf(fmaxf(y0f,        0.0f), (float)(CROP_H - 1));
    int y1 = (int)fminf(fmaxf(y0f + 1.0f, 0.0f), (float)(CROP_H - 1));

    // --- async-stage pooled rows y0,y1 (cols 50..299) for all 3 channels ---
    // smem layout: [channel][row(0/1)][256 floats, col-50 at index 0..249]
    __shared__ float smem[6 * 256];
    unsigned lds_base = (unsigned)(uintptr_t)(&smem[0]);   // low 32b = LDS offset
    uint64_t gbase    = (uint64_t)(uintptr_t)pooled;

    #pragma unroll
    for (int it = 0; it < 4; ++it) {
        int i = tid + it * 416;                // 0..1499 staged elements
        if (i < 1500) {
            int ch   = i / 500;
            int rem2 = i - ch * 500;
            int r    = rem2 / 250;
            int col  = rem2 - r * 250;         // 0..249  -> pooled col col+50
            int row  = r ? y1 : y0;
            unsigned voff  = (unsigned)(((ch * PDIM + row) * PDIM + (col + CROP_X0)) * 4);
            unsigned laddr = lds_base + (unsigned)(((ch * 2 + r) * 256 + col) * 4);
            asm volatile("global_load_async_to_lds_b32 %0, %1, %2"
                         :: "v"(laddr), "v"(voff), "s"(gbase)
                         : "memory");
        }
    }
    asm volatile("s_wait_asynccnt 0" ::: "memory");
    __syncthreads();

    // --- per-pixel bilinear from LDS ---
    float xf  = (float)tid;
    float u   = (xf - L + 0.5f) * ((float)CROP_W / Wt) - 0.5f;
    float x0f = floorf(u);
    float wx  = u - x0f;
    int x0 = (int)fminf(fmaxf(x0f,        0.0f), (float)(CROP_W - 1));
    int x1 = (int)fminf(fmaxf(x0f + 1.0f, 0.0f), (float)(CROP_W - 1));
    float mx0 = (CROP_X0 + x0 < PDIM) ? 1.0f : 0.0f;
    float mx1 = (CROP_X0 + x1 < PDIM) ? 1.0f : 0.0f;
    int c0 = min(x0, PDIM - 1 - CROP_X0);      // LDS col index (sx-50)
    int c1 = min(x1, PDIM - 1 - CROP_X0);

    bool in_box = (xf >= L) && (xf < L + Wt);  // vertical part already true
    float w00 = (1.0f - wx) * (1.0f - wy) * mx0;
    float w01 = wx          * (1.0f - wy) * mx1;
    float w10 = (1.0f - wx) * wy          * mx0;
    float w11 = wx          * wy          * mx1;

    #pragma unroll
    for (int c = 0; c < 3; ++c) {
        const float* s0 = &smem[(c * 2 + 0) * 256];
        const float* s1 = &smem[(c * 2 + 1) * 256];
        float val = s0[c0] * w00 + s0[c1] * w01 + s1[c0] * w10 + s1[c1] * w11;
        orow[(size_t)c * (SIMG * SIMG) + tid] = in_box ? val : 0.0f;
    }
}

// ---------------------------------------------------------------------------
extern "C" void kernel_launch(void* const* d_in, const int* in_sizes, int n_in,
                              void* d_out, int out_size, void* d_ws, size_t ws_size,
                              hipStream_t stream) {
    const float* adv = (const float*)d_in[0];   // (3,300,300) f32
    const float* lab = (const float*)d_in[1];   // (8,8,39)    f32
    const int*   img = (const int*)  d_in[2];   // scalar 416
    float* out    = (float*)d_out;              // (8,8,3,416,416) f32
    float* pooled = (float*)d_ws;               // 3*300*300 f32 scratch

    int n_med = 3 * PDIM * PDIM;
    median_pool_kernel<<<(n_med + 255) / 256, 256, 0, stream>>>(adv, pooled);

    dim3 grid(64, SIMG, 1);
    patch_apply_kernel<<<grid, 416, 0, stream>>>(pooled, lab, img, out);
}